// GATModel_21328807592517
// MI455X (gfx1250) — compile-verified
//
#include <hip/hip_runtime.h>
#include <math.h>

// ---------------------------------------------------------------------------
// GAT model on gfx1250: 2x GATConv(8 heads x 32) + ELU + mean pool + linear.
// GEMMs run on V_WMMA_F32_16X16X4_F32 (full fp32 WMMA), A-tile staged in LDS
// and shared by all 16 waves of the block. Aggregation is wave-per-edge
// coalesced gather + global_atomic_add_f32 scatter.
// ---------------------------------------------------------------------------

typedef float v2f __attribute__((ext_vector_type(2)));
typedef float v8f __attribute__((ext_vector_type(8)));

#define F_IN   128
#define D1     256     // heads * hid
#define HEADS  8
#define HID    32
#define NEG_SLOPE 0.2f

// ---------------- utility -----------------

__global__ void fill_f32(float* __restrict__ p, float v, int n) {
    int t = blockIdx.x * blockDim.x + threadIdx.x;
    if (t < n) p[t] = v;
}

// atomic float max via int/uint ordering trick (buffer initialized to -inf)
__device__ __forceinline__ void atomicMaxF(float* addr, float val) {
    if (val >= 0.0f) atomicMax((int*)addr, __float_as_int(val));
    else             atomicMin((unsigned int*)addr, __float_as_uint(val));
}

// ---------------- fp32 WMMA GEMM: C[M,256] = A[M,K] @ B[K,256] --------------
// One block (16 waves, 512 threads) per 16-row output panel. The 16xK A tile
// is staged once in LDS (padded rows: bank-conflict-free b64 reads) and
// shared by all waves; wave w computes output tile (blockIdx.x, tn=w) with
// v_wmma_f32_16x16x4_f32, stepping K by 4.
template<int K>
__global__ void gemm_wmma_f32(const float* __restrict__ A,
                              const float* __restrict__ B,
                              float* __restrict__ C) {
    constexpr int NC    = D1;
    constexpr int LDSTR = K + 4;            // row pad: lane bank step = 4
    __shared__ float ldsA[16 * LDSTR];

    const int tm   = blockIdx.x;
    const int tid  = threadIdx.x;           // 0..511
    const int tn   = tid >> 5;              // wave id = output col tile
    const int lane = tid & 31;

    // cooperative float4 fill of the 16xK A tile
    const float4* Ag = (const float4*)(A + (size_t)tm * 16 * K);
    for (int i = tid; i < 16 * (K / 4); i += 512) {
        int row = i / (K / 4), c4 = i % (K / 4);
        *(float4*)(ldsA + row * LDSTR + c4 * 4) = Ag[(size_t)row * (K / 4) + c4];
    }
    __syncthreads();

    const int r     = lane & 15;            // row (A) / col (B) within tile
    const int khalf = (lane >> 4) << 1;     // 0 for lanes 0-15, 2 for 16-31
    const float* arow = ldsA + r * LDSTR + khalf;
    const float* bcol = B + tn * 16 + r;

    v8f c = {0.f, 0.f, 0.f, 0.f, 0.f, 0.f, 0.f, 0.f};
    #pragma unroll 8
    for (int k0 = 0; k0 < K; k0 += 4) {
        v2f a, b;
        // A 16x4 fragment: lane holds A[M=r][k0+khalf], A[M=r][k0+khalf+1]
        a.x = arow[k0];
        a.y = arow[k0 + 1];
        // B 4x16 fragment (mirror): lane holds B[k0+khalf][col=r], B[k0+khalf+1][col=r]
        b.x = bcol[(k0 + khalf) * NC];
        b.y = bcol[(k0 + khalf + 1) * NC];
        c = __builtin_amdgcn_wmma_f32_16x16x4_f32(
                /*neg_a=*/false, a, /*neg_b=*/false, b,
                /*c_mod=*/(short)0, c, /*reuse_a=*/false, /*reuse_b=*/false);
    }

    // C/D layout: VGPR v, lane l -> M = v + 8*(l>=16), N = l&15
    float* crow = C + (size_t)(tm * 16 + ((lane >> 4) << 3)) * NC + tn * 16 + r;
    #pragma unroll
    for (int v = 0; v < 8; ++v) crow[v * NC] = c[v];
}

// --------- per-node attention logits: a[n,h] = sum_c h[n,h,c]*att[h,c] ------
__global__ void node_alpha(const float* __restrict__ h,
                           const float* __restrict__ att_src,
                           const float* __restrict__ att_dst,
                           float* __restrict__ asrc,
                           float* __restrict__ adst, int N) {
    int t = blockIdx.x * blockDim.x + threadIdx.x;
    if (t >= N * HEADS) return;
    int n = t >> 3, hh = t & 7;
    const float4* hp = (const float4*)(h + (size_t)n * D1 + hh * HID);
    const float4* as = (const float4*)(att_src + hh * HID);
    const float4* ad = (const float4*)(att_dst + hh * HID);
    float s1 = 0.f, s2 = 0.f;
    #pragma unroll
    for (int i = 0; i < HID / 4; ++i) {
        float4 hv = hp[i], a = as[i], b = ad[i];
        s1 += hv.x * a.x + hv.y * a.y + hv.z * a.z + hv.w * a.w;
        s2 += hv.x * b.x + hv.y * b.y + hv.z * b.z + hv.w * b.w;
    }
    asrc[t] = s1;
    adst[t] = s2;
}

// --------- edge pass 1: e = leaky_relu(asrc[src]+adst[dst]); seg-max --------
__global__ void edge_score(const int* __restrict__ src, const int* __restrict__ dst,
                           const float* __restrict__ asrc, const float* __restrict__ adst,
                           float* __restrict__ edge_e, float* __restrict__ nmax, int E) {
    int t = blockIdx.x * blockDim.x + threadIdx.x;
    if (t >= E * HEADS) return;
    int e = t >> 3, hh = t & 7;
    int s = src[e], d = dst[e];
    float v = asrc[s * HEADS + hh] + adst[d * HEADS + hh];
    v = (v > 0.f) ? v : NEG_SLOPE * v;
    edge_e[t] = v;
    atomicMaxF(&nmax[d * HEADS + hh], v);
}

// --------- edge pass 2: p = exp(e - max[dst]); seg-sum ----------------------
__global__ void edge_exp(const int* __restrict__ dst,
                         float* __restrict__ edge_e,
                         const float* __restrict__ nmax,
                         float* __restrict__ nsum, int E) {
    int t = blockIdx.x * blockDim.x + threadIdx.x;
    if (t >= E * HEADS) return;
    int e = t >> 3, hh = t & 7;
    int d = dst[e];
    float p = expf(edge_e[t] - nmax[d * HEADS + hh]);
    edge_e[t] = p;
    atomicAdd(&nsum[d * HEADS + hh], p);
}

// --------- edge pass 3: out[dst] += h[src] * p/(sum[dst]+eps) ---------------
// One wave per edge; lane l handles channels [8l, 8l+8) (all in head l/4).
__global__ void gat_aggregate(const float* __restrict__ h,
                              const float* __restrict__ edge_p,
                              const float* __restrict__ nsum,
                              const int* __restrict__ src, const int* __restrict__ dst,
                              float* __restrict__ out, int E) {
    int wave = (blockIdx.x * blockDim.x + threadIdx.x) >> 5;
    int lane = threadIdx.x & 31;
    if (wave >= E) return;
    int s = src[wave], d = dst[wave];
    int head = lane >> 2;
    float p     = edge_p[(size_t)wave * HEADS + head];
    float den   = nsum[(size_t)d * HEADS + head] + 1e-16f;
    float alpha = p / den;
    const float4* hp = (const float4*)(h + (size_t)s * D1 + lane * 8);
    float4 v0 = hp[0], v1 = hp[1];
    float* op = out + (size_t)d * D1 + lane * 8;
    atomicAdd(op + 0, v0.x * alpha);
    atomicAdd(op + 1, v0.y * alpha);
    atomicAdd(op + 2, v0.z * alpha);
    atomicAdd(op + 3, v0.w * alpha);
    atomicAdd(op + 4, v1.x * alpha);
    atomicAdd(op + 5, v1.y * alpha);
    atomicAdd(op + 6, v1.z * alpha);
    atomicAdd(op + 7, v1.w * alpha);
}

// --------- out = elu(out + bias) (in place) ---------------------------------
__global__ void bias_elu(float* __restrict__ buf, const float* __restrict__ bias, int total) {
    int t = blockIdx.x * blockDim.x + threadIdx.x;
    if (t >= total) return;
    float v = buf[t] + bias[t & (D1 - 1)];
    buf[t] = (v > 0.f) ? v : (expf(v) - 1.0f);
}

// --------- column sums for mean pooling (256 blocks x 256 threads) ----------
__global__ void colsum(const float* __restrict__ buf, float* __restrict__ g, int N) {
    int c = threadIdx.x;
    float acc = 0.f;
    for (int n = blockIdx.x; n < N; n += gridDim.x)
        acc += buf[(size_t)n * D1 + c];
    atomicAdd(&g[c], acc);
}

// --------- final: out = (g/N) . lin_w[0:256] + u*lw[256] + w*lw[257] + lb ---
__global__ void finalize(const float* __restrict__ g, const float* __restrict__ linw,
                         const float* __restrict__ linb, const float* __restrict__ u,
                         const float* __restrict__ w, float* __restrict__ outp, int N) {
    __shared__ float sh[D1];
    int t = threadIdx.x;
    sh[t] = g[t] * linw[t];
    __syncthreads();
    for (int s = D1 / 2; s > 0; s >>= 1) {
        if (t < s) sh[t] += sh[t + s];
        __syncthreads();
    }
    if (t == 0)
        outp[0] = sh[0] / (float)N + u[0] * linw[D1] + w[0] * linw[D1 + 1] + linb[0];
}

// ---------------------------------------------------------------------------

static inline int cdiv(long long a, long long b) { return (int)((a + b - 1) / b); }

extern "C" void kernel_launch(void* const* d_in, const int* in_sizes, int n_in,
                              void* d_out, int out_size, void* d_ws, size_t ws_size,
                              hipStream_t stream) {
    const float* x        = (const float*)d_in[0];
    const int*   eidx     = (const int*)  d_in[1];
    const float* u        = (const float*)d_in[2];
    const float* w        = (const float*)d_in[3];
    const float* W1       = (const float*)d_in[4];
    const float* att_src1 = (const float*)d_in[5];
    const float* att_dst1 = (const float*)d_in[6];
    const float* bias1    = (const float*)d_in[7];
    const float* W2       = (const float*)d_in[8];
    const float* att_src2 = (const float*)d_in[9];
    const float* att_dst2 = (const float*)d_in[10];
    const float* bias2    = (const float*)d_in[11];
    const float* lin_w    = (const float*)d_in[12];
    const float* lin_b    = (const float*)d_in[13];

    const int N = in_sizes[0] / F_IN;
    const int E = in_sizes[1] / 2;
    const int* src = eidx;
    const int* dst = eidx + E;

    // workspace layout (floats)
    float* W_ = (float*)d_ws;
    size_t o = 0;
    float* bufH   = W_ + o; o += (size_t)N * D1;     // h1, later h2
    float* bufO   = W_ + o; o += (size_t)N * D1;     // o1, later o2 (accumulator)
    float* asrc   = W_ + o; o += (size_t)N * HEADS;
    float* adst   = W_ + o; o += (size_t)N * HEADS;
    float* nmax   = W_ + o; o += (size_t)N * HEADS;
    float* nsum   = W_ + o; o += (size_t)N * HEADS;
    float* edge_p = W_ + o; o += (size_t)E * HEADS;
    float* gvec   = W_ + o; o += D1;

    const float NEG_INF = -INFINITY;
    const int TPB = 256;
    const int gemmBlocks = N / 16;                    // one 16-row panel per block
    const int ehBlocks   = cdiv((long long)E * HEADS, TPB);
    const int nhBlocks   = cdiv((long long)N * HEADS, TPB);
    const int ndBlocks   = cdiv((long long)N * D1, TPB);
    const int aggBlocks  = cdiv(E, TPB / 32);

    // ---------------- layer 1 ----------------
    fill_f32<<<ndBlocks, TPB, 0, stream>>>(bufO, 0.f, N * D1);
    fill_f32<<<nhBlocks, TPB, 0, stream>>>(nsum, 0.f, N * HEADS);
    fill_f32<<<nhBlocks, TPB, 0, stream>>>(nmax, NEG_INF, N * HEADS);

    gemm_wmma_f32<F_IN><<<gemmBlocks, 512, 0, stream>>>(x, W1, bufH);
    node_alpha<<<nhBlocks, TPB, 0, stream>>>(bufH, att_src1, att_dst1, asrc, adst, N);
    edge_score<<<ehBlocks, TPB, 0, stream>>>(src, dst, asrc, adst, edge_p, nmax, E);
    edge_exp<<<ehBlocks, TPB, 0, stream>>>(dst, edge_p, nmax, nsum, E);
    gat_aggregate<<<aggBlocks, TPB, 0, stream>>>(bufH, edge_p, nsum, src, dst, bufO, E);
    bias_elu<<<ndBlocks, TPB, 0, stream>>>(bufO, bias1, N * D1);

    // ---------------- layer 2 ----------------
    gemm_wmma_f32<D1><<<gemmBlocks, 512, 0, stream>>>(bufO, W2, bufH);
    // bufO (o1) no longer needed -> reuse as layer-2 output accumulator
    fill_f32<<<ndBlocks, TPB, 0, stream>>>(bufO, 0.f, N * D1);
    fill_f32<<<nhBlocks, TPB, 0, stream>>>(nsum, 0.f, N * HEADS);
    fill_f32<<<nhBlocks, TPB, 0, stream>>>(nmax, NEG_INF, N * HEADS);

    node_alpha<<<nhBlocks, TPB, 0, stream>>>(bufH, att_src2, att_dst2, asrc, adst, N);
    edge_score<<<ehBlocks, TPB, 0, stream>>>(src, dst, asrc, adst, edge_p, nmax, E);
    edge_exp<<<ehBlocks, TPB, 0, stream>>>(dst, edge_p, nmax, nsum, E);
    gat_aggregate<<<aggBlocks, TPB, 0, stream>>>(bufH, edge_p, nsum, src, dst, bufO, E);
    bias_elu<<<ndBlocks, TPB, 0, stream>>>(bufO, bias2, N * D1);

    // ---------------- readout ----------------
    fill_f32<<<1, D1, 0, stream>>>(gvec, 0.f, D1);
    colsum<<<256, D1, 0, stream>>>(bufO, gvec, N);
    finalize<<<1, D1, 0, stream>>>(gvec, lin_w, lin_b, u, w, (float*)d_out, N);
}